// SyntheticBlock_4063039062082
// MI455X (gfx1250) — compile-verified
//
#include <hip/hip_runtime.h>
#include <math.h>

typedef __attribute__((ext_vector_type(2))) float v2f;
typedef __attribute__((ext_vector_type(8))) float v8f;

constexpr int N_ = 50000;
constexpr int C_ = 128;
constexpr int E_ = 800000;
constexpr int ROW_TILES    = N_ / 16;  // 3125 (exact)
constexpr int EDGE_TILES32 = E_ / 32;  // 25000 (exact)
constexpr int KS = 132;                // K-major LDS stride (bank-friendly, 8B-aligned pairs)
constexpr unsigned ENC_NEGINF = 0x007FFFFFu;  // order-encoded -inf

__device__ __forceinline__ float lrelu(float v) { return fmaxf(v, 0.01f * v); }

// Order-preserving float->uint encoding so segment_max can use u32 atomic max.
__device__ __forceinline__ unsigned fenc(float f) {
  unsigned u = __float_as_uint(f);
  return (u & 0x80000000u) ? ~u : (u | 0x80000000u);
}
__device__ __forceinline__ float fdec(unsigned u) {
  unsigned b = (u & 0x80000000u) ? (u & 0x7FFFFFFFu) : ~u;
  return __uint_as_float(b);
}

// One V_WMMA_F32_16X16X4_F32 step. Both operands are contiguous (k,k+1) pairs
// in LDS (A: row-major-K tile; B: K-transposed weights) -> single ds_load_b64.
__device__ __forceinline__ v8f wmma_step(const float* a_ptr, const float* b_ptr,
                                         v8f acc) {
  v2f a = *(const v2f*)a_ptr;
  v2f b = *(const v2f*)b_ptr;
  return __builtin_amdgcn_wmma_f32_16x16x4_f32(false, a, false, b,
                                               (short)0, acc, false, false);
}

// ---------------------------------------------------------------------------
// Init: agg keys = enc(-inf); channel-stat accumulators = 0
__global__ void k_init(unsigned* __restrict__ aggK, float* __restrict__ sums) {
  const size_t idx = (size_t)blockIdx.x * blockDim.x + threadIdx.x;
  if (idx < (size_t)N_ * C_) aggK[idx] = ENC_NEGINF;
  if (idx < 256) sums[idx] = 0.0f;
}

// ---------------------------------------------------------------------------
// Generic (N,128)@(128,128) GEMM, K-transposed weight in LDS, 16-row tiles.
// mode 0: Y = lrelu(A@W + b)
// mode 1: A = decode(aggK) with enc(-inf)->0 ; Y = lrelu(A@W + b)
// mode 2: Y = lrelu(A@W + b + addH + noise*ns)   (in-place over A is safe)
__global__ __launch_bounds__(256) void k_gemm(
    const float* __restrict__ A, const unsigned* __restrict__ Akeys,
    const float* __restrict__ W, const float* __restrict__ bias,
    float* __restrict__ Y, const float* __restrict__ addH,
    const float* __restrict__ noise, const float* __restrict__ nsPtr,
    int mode) {
  extern __shared__ float smem[];
  float* Wt = smem;             // 128*KS  (K-transposed: Wt[n*KS + k])
  float* bS = Wt + 128 * KS;    // 128
  float* As = bS + 128;         // 16*KS
  const int tid = threadIdx.x;
  for (int i = tid; i < 128 * 128; i += 256) {
    const int k = i >> 7, n = i & 127;
    Wt[n * KS + k] = W[i];
  }
  if (tid < 128) bS[tid] = bias[tid];
  const float nsv = (mode == 2) ? nsPtr[0] : 0.0f;
  __syncthreads();

  const int lane = tid & 31, l15 = lane & 15, hi = lane >> 4;
  const int n0 = (tid >> 5) * 16;
  const float* ap = As + l15 * KS + 2 * hi;
  const float* bp = Wt + (n0 + l15) * KS + 2 * hi;

  for (int t = blockIdx.x; t < ROW_TILES; t += gridDim.x) {
    const int r0 = t * 16;
    for (int i = tid; i < 16 * 128; i += 256) {
      const int r = i >> 7, c = i & 127;
      float v;
      if (mode == 1) {
        const unsigned u = Akeys[(size_t)(r0 + r) * C_ + c];
        v = (u == ENC_NEGINF) ? 0.0f : fdec(u);
      } else {
        v = A[(size_t)(r0 + r) * C_ + c];
      }
      As[r * KS + c] = v;
    }
    // Prefetch next tile's A rows (8 KB) while this tile computes.
    const int tn = t + gridDim.x;
    if (tn < ROW_TILES) {
      const char* np = (mode == 1) ? (const char*)(Akeys + (size_t)tn * 16 * C_)
                                   : (const char*)(A + (size_t)tn * 16 * C_);
      __builtin_prefetch(np + tid * 32, 0, 1);
    }
    __syncthreads();

    // 4 independent accumulator chains over K (hides WMMA pipe latency).
    v8f a0 = {0,0,0,0,0,0,0,0}, a1 = a0, a2 = a0, a3 = a0;
#pragma unroll
    for (int k4 = 0; k4 < 128; k4 += 16) {
      a0 = wmma_step(ap + k4,      bp + k4,      a0);
      a1 = wmma_step(ap + k4 + 4,  bp + k4 + 4,  a1);
      a2 = wmma_step(ap + k4 + 8,  bp + k4 + 8,  a2);
      a3 = wmma_step(ap + k4 + 12, bp + k4 + 12, a3);
    }
    v8f acc = (a0 + a1) + (a2 + a3);
#pragma unroll
    for (int v = 0; v < 8; ++v) {
      const int row = v + 8 * hi;
      const int col = n0 + l15;
      const size_t gi = (size_t)(r0 + row) * C_ + col;
      float val = acc[v] + bS[col];
      if (mode == 2) val += addH[gi] + noise[gi] * nsv;
      Y[gi] = lrelu(val);
    }
    __syncthreads();
  }
}

// ---------------------------------------------------------------------------
// delta = tanh(t @ Wh2 + bh2), Wh2 is (128,3). One wave per node.
__global__ __launch_bounds__(256) void k_delta(
    const float* __restrict__ t, const float* __restrict__ Wh2,
    const float* __restrict__ bh2, float* __restrict__ delta) {
  const int lane = threadIdx.x & 31;
  const int node = blockIdx.x * 8 + (threadIdx.x >> 5);
  if (node >= N_) return;
  float a0 = 0.f, a1 = 0.f, a2 = 0.f;
  for (int k = lane; k < 128; k += 32) {
    const float v = t[(size_t)node * C_ + k];
    a0 += v * Wh2[k * 3 + 0];
    a1 += v * Wh2[k * 3 + 1];
    a2 += v * Wh2[k * 3 + 2];
  }
  for (int off = 16; off > 0; off >>= 1) {
    a0 += __shfl_xor(a0, off);
    a1 += __shfl_xor(a1, off);
    a2 += __shfl_xor(a2, off);
  }
  if (lane == 0) {
    delta[node * 3 + 0] = tanhf(a0 + bh2[0]);
    delta[node * 3 + 1] = tanhf(a1 + bh2[1]);
    delta[node * 3 + 2] = tanhf(a2 + bh2[2]);
  }
}

// ---------------------------------------------------------------------------
// Fused edge MLP + segment max, 32-edge tiles (two 16-row sub-tiles share
// every B fragment -> 1.5 b64 LDS loads per WMMA, 4 independent chains):
//  x = [pos[src]-pos[dst]+delta[dst], h[src]] (131, padded to 132)
//  e = lrelu(x @ Wf + bf);   agg[dst] = max(agg[dst], e)  via u32 atomic max
__global__ __launch_bounds__(256) void k_edge(
    const float* __restrict__ h, const float* __restrict__ pos,
    const float* __restrict__ delta, const int* __restrict__ ei,
    const float* __restrict__ Wf, const float* __restrict__ bf,
    unsigned* __restrict__ aggK) {
  extern __shared__ float smem[];
  float* Wft = smem;              // 128*KS, K-transposed, rows 131 zero-padded
  float* bS  = Wft + 128 * KS;    // 128
  float* Xs  = bS + 128;          // 32*KS
  int* srcs  = (int*)(Xs + 32 * KS); // 32
  int* dsts  = srcs + 32;            // 32
  const int tid = threadIdx.x;
  for (int i = tid; i < 132 * 128; i += 256) {
    const int k = i >> 7, n = i & 127;
    Wft[n * KS + k] = (k < 131) ? Wf[k * 128 + n] : 0.0f;
  }
  if (tid < 128) bS[tid] = bf[tid];
  __syncthreads();

  const int lane = tid & 31, l15 = lane & 15, hi = lane >> 4;
  const int n0 = (tid >> 5) * 16;
  const float* ap0 = Xs + l15 * KS + 2 * hi;        // sub-tile 0: edges 0..15
  const float* ap1 = ap0 + 16 * KS;                 // sub-tile 1: edges 16..31
  const float* bp  = Wft + (n0 + l15) * KS + 2 * hi;

  for (int t = blockIdx.x; t < EDGE_TILES32; t += gridDim.x) {
    const int e0 = t * 32;
    if (tid < 32) {
      srcs[tid] = ei[e0 + tid];        // edge_index[0] = src
      dsts[tid] = ei[E_ + e0 + tid];   // edge_index[1] = dst
    }
    const int tn = t + gridDim.x;
    if (tid == 0 && tn < EDGE_TILES32) {
      __builtin_prefetch((const char*)(ei + tn * 32), 0, 1);
      __builtin_prefetch((const char*)(ei + E_ + tn * 32), 0, 1);
    }
    __syncthreads();
    for (int i = tid; i < 32 * KS; i += 256) {
      const int r = i / KS, c = i - r * KS;
      const int s = srcs[r], d = dsts[r];
      float v;
      if (c < 3)        v = pos[s * 3 + c] - pos[d * 3 + c] + delta[d * 3 + c];
      else if (c < 131) v = h[(size_t)s * C_ + (c - 3)];
      else              v = 0.0f;
      Xs[i] = v;
    }
    __syncthreads();

    // acc{m}{p}: m = sub-tile, p = even/odd K-step chain. B shared across m.
    v8f c00 = {0,0,0,0,0,0,0,0}, c01 = c00, c10 = c00, c11 = c00;
#pragma unroll
    for (int k8 = 0; k8 < 128; k8 += 8) {
      c00 = wmma_step(ap0 + k8,     bp + k8,     c00);
      c10 = wmma_step(ap1 + k8,     bp + k8,     c10);
      c01 = wmma_step(ap0 + k8 + 4, bp + k8 + 4, c01);
      c11 = wmma_step(ap1 + k8 + 4, bp + k8 + 4, c11);
    }
    c00 = wmma_step(ap0 + 128, bp + 128, c00);  // K tail: 128..131
    c10 = wmma_step(ap1 + 128, bp + 128, c10);
    v8f acc0 = c00 + c01;
    v8f acc1 = c10 + c11;
#pragma unroll
    for (int v = 0; v < 8; ++v) {
      const int row = v + 8 * hi;
      const int col = n0 + l15;
      const float y0 = lrelu(acc0[v] + bS[col]);
      const float y1 = lrelu(acc1[v] + bS[col]);
      atomicMax(&aggK[(size_t)dsts[row] * C_ + col], fenc(y0));
      atomicMax(&aggK[(size_t)dsts[16 + row] * C_ + col], fenc(y1));
    }
    __syncthreads();
  }
}

// ---------------------------------------------------------------------------
// Per-channel sum / sum-of-squares of hn (coalesced row-pair strides).
__global__ __launch_bounds__(256) void k_stats(const float* __restrict__ hn,
                                               float* __restrict__ sums) {
  const int tid = threadIdx.x;
  const int c = tid & 127;
  const int half = tid >> 7;
  float s = 0.0f, s2 = 0.0f;
  for (int r = blockIdx.x * 2 + half; r < N_; r += gridDim.x * 2) {
    const float v = hn[(size_t)r * C_ + c];
    s += v;
    s2 += v * v;
  }
  __shared__ float sh[512];
  sh[tid] = s;
  sh[256 + tid] = s2;
  __syncthreads();
  if (tid < 128) {
    atomicAdd(&sums[c], sh[tid] + sh[tid + 128]);
    atomicAdd(&sums[128 + c], sh[256 + tid] + sh[256 + tid + 128]);
  }
}

__global__ void k_mv(const float* __restrict__ sums, float* __restrict__ mv) {
  const int c = threadIdx.x;
  if (c < 128) {
    const float inv_n = 1.0f / (float)N_;
    const float mean = sums[c] * inv_n;
    const float var = sums[128 + c] * inv_n - mean * mean;  // population var
    mv[c] = mean;
    mv[128 + c] = rsqrtf(var + 1e-5f);
  }
}

// ---------------------------------------------------------------------------
// gb = style @ Wa + ba ; out = gamma * (hn - mean) * invstd + beta
// Each wave computes gamma (cols n0..) and beta (cols 128+n0..) tiles sharing
// one A fragment -> 4 independent WMMA chains, epilogue stays lane-local.
__global__ __launch_bounds__(256) void k_out(
    const float* __restrict__ style, const float* __restrict__ Wa,
    const float* __restrict__ ba, const float* __restrict__ hn,
    const float* __restrict__ mv, float* __restrict__ out) {
  extern __shared__ float smem[];
  float* Wat = smem;              // 256*KS, K-transposed: Wat[n*KS + k]
  float* baS = Wat + 256 * KS;    // 256
  float* Ss  = baS + 256;         // 16*KS
  const int tid = threadIdx.x;
  for (int i = tid; i < 128 * 256; i += 256) {
    const int k = i >> 8, n = i & 255;
    Wat[n * KS + k] = Wa[i];
  }
  if (tid < 256) baS[tid] = ba[tid];
  __syncthreads();

  const int lane = tid & 31, l15 = lane & 15, hi = lane >> 4;
  const int n0 = (tid >> 5) * 16;
  const float* ap = Ss + l15 * KS + 2 * hi;
  const float* bg = Wat + (n0 + l15) * KS + 2 * hi;
  const float* bb = Wat + (128 + n0 + l15) * KS + 2 * hi;

  for (int t = blockIdx.x; t < ROW_TILES; t += gridDim.x) {
    const int r0 = t * 16;
    for (int i = tid; i < 16 * 128; i += 256) {
      const int r = i >> 7, c = i & 127;
      Ss[r * KS + c] = style[(size_t)(r0 + r) * C_ + c];
    }
    const int tn = t + gridDim.x;
    if (tn < ROW_TILES) {
      __builtin_prefetch((const char*)(style + (size_t)tn * 16 * C_) + tid * 32, 0, 1);
    }
    __syncthreads();

    v8f g0 = {0,0,0,0,0,0,0,0}, g1 = g0, b0 = g0, b1 = g0;
#pragma unroll
    for (int k4 = 0; k4 < 128; k4 += 8) {
      g0 = wmma_step(ap + k4,     bg + k4,     g0);
      b0 = wmma_step(ap + k4,     bb + k4,     b0);
      g1 = wmma_step(ap + k4 + 4, bg + k4 + 4, g1);
      b1 = wmma_step(ap + k4 + 4, bb + k4 + 4, b1);
    }
    v8f accG = g0 + g1;
    v8f accB = b0 + b1;
#pragma unroll
    for (int v = 0; v < 8; ++v) {
      const int row = v + 8 * hi;
      const int col = n0 + l15;
      const size_t gi = (size_t)(r0 + row) * C_ + col;
      const float gamma = accG[v] + baS[col];
      const float beta  = accB[v] + baS[128 + col];
      const float x = hn[gi];
      out[gi] = gamma * (x - mv[col]) * mv[128 + col] + beta;
    }
    __syncthreads();
  }
}

// ---------------------------------------------------------------------------
extern "C" void kernel_launch(void* const* d_in, const int* in_sizes, int n_in,
                              void* d_out, int out_size, void* d_ws, size_t ws_size,
                              hipStream_t stream) {
  (void)in_sizes; (void)n_in; (void)out_size; (void)ws_size;
  const float* h     = (const float*)d_in[0];
  const float* pos   = (const float*)d_in[1];
  const float* style = (const float*)d_in[2];
  const int*   ei    = (const int*)d_in[3];
  const float* Wh1   = (const float*)d_in[4];
  const float* bh1   = (const float*)d_in[5];
  const float* Wh2   = (const float*)d_in[6];
  const float* bh2   = (const float*)d_in[7];
  const float* Wf    = (const float*)d_in[8];
  const float* bf    = (const float*)d_in[9];
  const float* Wg1   = (const float*)d_in[10];
  const float* bg1   = (const float*)d_in[11];
  const float* Wg2   = (const float*)d_in[12];
  const float* bg2   = (const float*)d_in[13];
  const float* Wa    = (const float*)d_in[14];
  const float* ba    = (const float*)d_in[15];
  const float* ns    = (const float*)d_in[16];
  const float* noise = (const float*)d_in[17];
  float* out = (float*)d_out;
  float* ws  = (float*)d_ws;

  const size_t NC = (size_t)N_ * C_;
  float*    buf0 = ws;                       // t -> t2 -> hn  (N*C)
  unsigned* aggK = (unsigned*)(ws + NC);     // encoded segment-max keys (N*C)
  float*    dlt  = ws + 2 * NC;              // delta (N*3)
  float*    sums = dlt + (size_t)3 * N_;     // 256 (sum, sumsq)
  float*    mv   = sums + 256;               // 256 (mean, invstd)

  const size_t smemGemm = (128 * KS + 128 + 16 * KS) * sizeof(float);
  const size_t smemEdge = (128 * KS + 128 + 32 * KS) * sizeof(float) + 64 * sizeof(int);
  const size_t smemOut  = (256 * KS + 256 + 16 * KS) * sizeof(float);

  k_init<<<(unsigned)((NC + 255) / 256), 256, 0, stream>>>(aggK, sums);
  // t = lrelu(h @ Wh1 + bh1)
  k_gemm<<<512, 256, smemGemm, stream>>>(h, nullptr, Wh1, bh1, buf0,
                                         nullptr, nullptr, nullptr, 0);
  // delta = tanh(t @ Wh2 + bh2)
  k_delta<<<(N_ + 7) / 8, 256, 0, stream>>>(buf0, Wh2, bh2, dlt);
  // fused edge MLP + segment max into aggK
  k_edge<<<2048, 256, smemEdge, stream>>>(h, pos, dlt, ei, Wf, bf, aggK);
  // t2 = lrelu(decode(agg) @ Wg1 + bg1)
  k_gemm<<<512, 256, smemGemm, stream>>>(nullptr, aggK, Wg1, bg1, buf0,
                                         nullptr, nullptr, nullptr, 1);
  // hn = lrelu(t2 @ Wg2 + bg2 + h + noise*ns)   (in-place)
  k_gemm<<<512, 256, smemGemm, stream>>>(buf0, nullptr, Wg2, bg2, buf0,
                                         h, noise, ns, 2);
  k_stats<<<256, 256, 0, stream>>>(buf0, sums);
  k_mv<<<1, 128, 0, stream>>>(sums, mv);
  // gb = style @ Wa + ba ; out = gamma*normed + beta
  k_out<<<512, 256, smemOut, stream>>>(style, Wa, ba, buf0, mv, out);
}